// DecoderModelWrapper_46935402611348
// MI455X (gfx1250) — compile-verified
//
#include <hip/hip_runtime.h>
#include <hip/hip_bf16.h>
#include <stdint.h>

// Problem constants (from the reference): L=8, B=2, H=8, S=4096, D=128
#define CL 8
#define CB 2
#define CH 8
#define CS 4096
#define CD 128
#define N_ELEM   ((long long)CL * CB * CH * CS * CD)   // 67,108,864 floats per tensor
#define N4_ELEM  (N_ELEM / 4)                          // 16,777,216 16B-chunks per tensor
#define MASK_ELEM (CB * CS)                            // 8192 floats

// CPol: gfx12-class encoding, TH in bits [2:0]; TH=1 -> non-temporal (streaming)
#define CPOL_NT 1

typedef float v4f __attribute__((ext_vector_type(4)));
typedef int   v4i __attribute__((vector_size(16)));    // matches builtin param type

// Address-space-qualified pointer element types for the async builtins
typedef __attribute__((address_space(1))) v4i as1_v4i;
typedef __attribute__((address_space(3))) v4i as3_v4i;

#ifndef __has_builtin
#define __has_builtin(x) 0
#endif

#if __has_builtin(__builtin_amdgcn_global_load_async_to_lds_b128) && \
    __has_builtin(__builtin_amdgcn_global_store_async_from_lds_b128)
#define USE_ASYNC_LDS 1
#else
#define USE_ASYNC_LDS 0
#endif

__device__ __forceinline__ void wait_asynccnt0() {
#if __has_builtin(__builtin_amdgcn_s_wait_asynccnt)
  __builtin_amdgcn_s_wait_asynccnt(0);
#else
  asm volatile("s_wait_asynccnt 0" ::: "memory");
#endif
}

// ---------------------------------------------------------------------------
// Kernel 1: mask = (attention_mask != 0) ? 1.0f : 0.0f   (B*S = 8192 elems)
// ---------------------------------------------------------------------------
__global__ __launch_bounds__(256) void mask_kernel(const int* __restrict__ am,
                                                   float* __restrict__ out) {
  int i = blockIdx.x * blockDim.x + threadIdx.x;
  if (i < MASK_ELEM) out[i] = (am[i] != 0) ? 1.0f : 0.0f;
}

// ---------------------------------------------------------------------------
// Kernel 2: bulk stream copy of k_cache and v_cache into the output.
// 1.07 GB total traffic -> HBM-bound, ~46 us floor at 23.3 TB/s.
// CDNA5 async global<->LDS B128 transfers (ASYNCcnt), non-temporal CPol,
// 8 outstanding B128 per lane per phase (32 KB LDS staging per block).
// ---------------------------------------------------------------------------
__global__ __launch_bounds__(256) void kv_bulk_copy(const v4i* __restrict__ k4,
                                                    const v4i* __restrict__ v4,
                                                    v4i* __restrict__ out4) {
  const long long total = 2LL * N4_ELEM;            // 33,554,432 chunks (mult of 2048)
#if USE_ASYNC_LDS
  __shared__ v4i stage[256 * 8];                    // 32 KB: 8 private B128 slots/thread
  const int tid = threadIdx.x;
  const long long block_span = 2048;                // chunks per block per round
  for (long long base = (long long)blockIdx.x * block_span; base < total;
       base += (long long)gridDim.x * block_span) {
#pragma unroll
    for (int j = 0; j < 8; ++j) {
      long long idx = base + (long long)j * 256 + tid;
      const v4i* src = (idx < N4_ELEM) ? (k4 + idx) : (v4 + (idx - N4_ELEM));
      __builtin_amdgcn_global_load_async_to_lds_b128(
          (as1_v4i*)src, (as3_v4i*)&stage[j * 256 + tid], 0, CPOL_NT);
    }
    wait_asynccnt0();                               // loads complete -> LDS valid
#pragma unroll
    for (int j = 0; j < 8; ++j) {
      long long idx = base + (long long)j * 256 + tid;
      __builtin_amdgcn_global_store_async_from_lds_b128(
          (as1_v4i*)(out4 + idx), (as3_v4i*)&stage[j * 256 + tid], 0, CPOL_NT);
    }
    wait_asynccnt0();                               // stores done -> slots reusable
  }
#else
  // Fallback: nontemporal B128 load/store through VGPRs (stream > L2, bypass).
  long long stride = (long long)gridDim.x * blockDim.x;
  for (long long i = (long long)blockIdx.x * blockDim.x + threadIdx.x; i < total;
       i += stride) {
    v4i val = (i < N4_ELEM) ? __builtin_nontemporal_load(k4 + i)
                            : __builtin_nontemporal_load(v4 + (i - N4_ELEM));
    __builtin_nontemporal_store(val, out4 + i);
  }
#endif
}

// ---------------------------------------------------------------------------
// Kernel 3: overwrite the selected row s == position_ids[b] for every (l,b,h)
// in both tensors with new_k / new_v. 2*L*B*H = 256 rows of 128 floats
// = 8192 16B stores total. Runs after kv_bulk_copy on the same stream.
// ---------------------------------------------------------------------------
__global__ __launch_bounds__(256) void kv_scatter(const v4f* __restrict__ nk4,
                                                  const v4f* __restrict__ nv4,
                                                  const int* __restrict__ pos,
                                                  v4f* __restrict__ out4) {
  int t = blockIdx.x * blockDim.x + threadIdx.x;    // 0 .. 8191
  if (t >= 2 * CL * CB * CH * (CD / 4)) return;
  int d4    = t & 31;                               // 16B-chunk within the D=128 row
  int r     = t >> 5;                               // 0..255
  int which = r >> 7;                               // 0 = k, 1 = v
  int lbh   = r & 127;                              // l*B*H + b*H + h
  int b     = (lbh >> 3) & (CB - 1);
  int p     = pos[b];                               // position_ids is (B,1)
  const v4f* src = (which ? nv4 : nk4) + lbh * (CD / 4) + d4;
  long long row = (long long)lbh * CS + p;          // row within (L,B,H,S)
  out4[(long long)which * N4_ELEM + row * (CD / 4) + d4] = *src;
}

// ---------------------------------------------------------------------------
// d_in order: k_cache, v_cache, new_k, new_v, attention_mask, position_ids
// d_out: [mask (8192 f32)][k_upd (N f32)][v_upd (N f32)]
// ---------------------------------------------------------------------------
extern "C" void kernel_launch(void* const* d_in, const int* in_sizes, int n_in,
                              void* d_out, int out_size, void* d_ws, size_t ws_size,
                              hipStream_t stream) {
  const v4i* k4  = (const v4i*)d_in[0];
  const v4i* v4  = (const v4i*)d_in[1];
  const v4f* nk4 = (const v4f*)d_in[2];
  const v4f* nv4 = (const v4f*)d_in[3];
  const int* am  = (const int*)d_in[4];
  const int* pos = (const int*)d_in[5];
  float* out = (float*)d_out;
  v4i* outkv4 = (v4i*)(out + MASK_ELEM);            // 32 KB in, 16B-aligned

  mask_kernel<<<(MASK_ELEM + 255) / 256, 256, 0, stream>>>(am, out);
  kv_bulk_copy<<<4096, 256, 0, stream>>>(k4, v4, outkv4);
  kv_scatter<<<32, 256, 0, stream>>>(nk4, nv4, pos, (v4f*)outkv4);
}